// AdaptiveAttention_64785286692947
// MI455X (gfx1250) — compile-verified
//
#include <hip/hip_runtime.h>

// ---------------------------------------------------------------------------
// AdaptiveAttention fused implementation for gfx1250 (MI455X, wave32, WMMA).
//
// Shapes: B=64, R=49, T=256, H=512, K=64.
// d_in: V[B,R,H], h_t[B,T,H], s_t[B,T,H], Wv[K,H], Wg[K,H], Wh[1,K], Ws[K,H]
// d_out: c_hat[B,T,H] ++ alpha[B,T,R] ++ beta[B,T,1]  (flat f32)
// ---------------------------------------------------------------------------

#define BB 64
#define RR 49
#define TT 256
#define HH 512
#define KK 64

typedef float v2f __attribute__((ext_vector_type(2)));
typedef float v8f __attribute__((ext_vector_type(8)));

// D = A(16x4 f32) * B(4x16 f32) + C(16x16 f32)  -- CDNA5 wave32 WMMA
#define WMMA_F32X4(a, b, c) \
    __builtin_amdgcn_wmma_f32_16x16x4_f32(false, (a), false, (b), (short)0, (c), false, false)

// Hardware transcendental tanh (gfx1250 V_TANH_F32). TRANS ops need one
// independent instruction before the result is consumed (co-execution hazard,
// ISA 7.x); the compiler cannot see through asm, so carry a v_nop inside.
__device__ __forceinline__ float fast_tanh(float x) {
    float y;
    asm volatile("v_tanh_f32 %0, %1\n\tv_nop" : "=v"(y) : "v"(x));
    return y;
}

// Async global->LDS copy (CDNA5, tracked by ASYNCcnt). VDST is a per-lane LDS
// byte offset = low 32 bits of the generic LDS address.
__device__ __forceinline__ void async_load_f32_to_lds(const float* gsrc, const float* ldst) {
    unsigned int lds_off = (unsigned int)(unsigned long long)ldst;
    asm volatile("global_load_async_to_lds_b32 %0, %1, off"
                 :: "v"(lds_off), "v"(gsrc) : "memory");
}
__device__ __forceinline__ void wait_asynccnt0() {
    asm volatile("s_wait_asynccnt 0x0" ::: "memory");
}

// ---------------------------------------------------------------------------
// Kernel 1: Y[m,n] = sum_k X[m,k] * W[n,k], K=512, N=64.
// 128 threads = 4 waves; each wave owns one 16-row M-tile and all 4 N-tiles.
// ---------------------------------------------------------------------------
__global__ __launch_bounds__(128) void proj_gemm_kernel(
    const float* __restrict__ X, const float* __restrict__ W,
    float* __restrict__ Y)
{
    const int lane  = threadIdx.x & 31;
    const int wave  = threadIdx.x >> 5;
    const int mtile = blockIdx.x * 4 + wave;
    const int mrow  = (mtile << 4) + (lane & 15);
    const int kh    = (lane >> 4) << 1;          // 0 or 2
    const int n     = lane & 15;

    const float* __restrict__ xrow = X + (size_t)mrow * HH;
    const float* __restrict__ w0 = W + (size_t)(n)      * HH;
    const float* __restrict__ w1 = W + (size_t)(n + 16) * HH;
    const float* __restrict__ w2 = W + (size_t)(n + 32) * HH;
    const float* __restrict__ w3 = W + (size_t)(n + 48) * HH;

    v8f acc0 = {}, acc1 = {}, acc2 = {}, acc3 = {};

    for (int k0 = 0; k0 < HH; k0 += 4) {
        const int k = k0 + kh;
        __builtin_prefetch(xrow + k + 128, 0, 3);   // near-scope prefetch, A stream
        v2f a  = *(const v2f*)(xrow + k);
        v2f b0 = *(const v2f*)(w0 + k);
        v2f b1 = *(const v2f*)(w1 + k);
        v2f b2 = *(const v2f*)(w2 + k);
        v2f b3 = *(const v2f*)(w3 + k);
        acc0 = WMMA_F32X4(a, b0, acc0);
        acc1 = WMMA_F32X4(a, b1, acc1);
        acc2 = WMMA_F32X4(a, b2, acc2);
        acc3 = WMMA_F32X4(a, b3, acc3);
    }

    // C/D layout: VGPR i -> M = i (lanes 0-15) or i+8 (lanes 16-31); N = lane&15.
    const int row0 = (mtile << 4) + ((lane >> 4) << 3);
    float* __restrict__ yb = Y + (size_t)row0 * KK + n;
    #pragma unroll
    for (int i = 0; i < 8; ++i) {
        yb[(size_t)i * KK +  0] = acc0[i];
        yb[(size_t)i * KK + 16] = acc1[i];
        yb[(size_t)i * KK + 32] = acc2[i];
        yb[(size_t)i * KK + 48] = acc3[i];
    }
}

// ---------------------------------------------------------------------------
// Kernel 2: fused tanh-score + dual softmax, register-resident softmax.
// Block = 256 threads (8 waves), one wave per t; 8 t's per block.
// gv[b] staged via async-to-LDS with 66-float pitch (bank = (2r+k)%64: no
// conflicts for b64 reads). (gh,wh) interleaved as float2 -> 1 ds_load_b64
// feeds both. Extension row r=49 uses raw gs with the same loop body since
// content_s = gs + gh. Softmax stats via __shfl_xor butterfly (no LDS).
// ---------------------------------------------------------------------------
#define GVP 66

__global__ __launch_bounds__(256) void attn_scores_kernel(
    const float* __restrict__ gv,   // [B][49][64]
    const float* __restrict__ gh,   // [B][256][64]
    const float* __restrict__ gs,   // [B][256][64]
    const float* __restrict__ Wh,   // [64]
    float* __restrict__ alpha_out,  // [B][256][49]
    float* __restrict__ beta_out)   // [B][256]
{
    __shared__ float  gvL[RR * GVP];
    __shared__ float2 ghwL[8][KK];   // (gh[k], wh[k]) per wave's t
    __shared__ float  gsL[8][KK];    // raw gs per wave's t

    const int b    = blockIdx.x >> 5;
    const int t0   = (blockIdx.x & 31) << 3;
    const int tid  = threadIdx.x;
    const int wave = tid >> 5;
    const int lane = tid & 31;
    const int t    = t0 + wave;

    // Block-level stage of gv[b] through the async-to-LDS path (ASYNCcnt).
    {
        const float* gvb = gv + (size_t)b * RR * KK;
        for (int i = tid; i < RR * KK; i += 256) {
            const int r = i >> 6, k = i & 63;
            async_load_f32_to_lds(gvb + i, &gvL[r * GVP + k]);
        }
    }
    // Per-wave stage of (gh,wh) and gs for this wave's t.
    {
        const size_t base = ((size_t)b * TT + t) * KK;
        ghwL[wave][lane]      = make_float2(gh[base + lane],      Wh[lane]);
        ghwL[wave][lane + 32] = make_float2(gh[base + lane + 32], Wh[lane + 32]);
        gsL[wave][lane]       = gs[base + lane];
        gsL[wave][lane + 32]  = gs[base + lane + 32];
    }
    wait_asynccnt0();
    __syncthreads();

    // Row assignment: row0 = lane (0..31); row1 = lane+32 (lanes 0..16 ->
    // gv rows 32..48, lane 17 -> extension row from gs, others dummy).
    const float* rowp0 = &gvL[lane * GVP];
    const float* rowp1 = (lane <= 16) ? &gvL[(lane + 32) * GVP]
                       : (lane == 17) ? &gsL[wave][0]
                                      : &gvL[0];
    const float2* __restrict__ ghw = &ghwL[wave][0];

    float acc0 = 0.0f, acc1 = 0.0f;
    #pragma unroll 4
    for (int k = 0; k < KK; k += 2) {
        const float2 p0 = ghw[k], p1 = ghw[k + 1];
        const v2f g0 = *(const v2f*)(rowp0 + k);
        const v2f g1 = *(const v2f*)(rowp1 + k);
        acc0 += fast_tanh(g0.x + p0.x) * p0.y;
        acc1 += fast_tanh(g1.x + p0.x) * p0.y;
        acc0 += fast_tanh(g0.y + p1.x) * p1.y;
        acc1 += fast_tanh(g1.y + p1.x) * p1.y;
    }

    // Register softmax over the 49 rows (alpha) and 50 rows (beta).
    const float NEG = -1e30f;
    const float z_hi = (lane <= 16) ? acc1 : NEG;        // rows 32..48
    const float zext = __shfl(acc1, 17, 32);             // extension score

    float m = fmaxf(acc0, z_hi);
    #pragma unroll
    for (int off = 16; off; off >>= 1) m = fmaxf(m, __shfl_xor(m, off, 32));

    const float e_lo = __expf(acc0 - m);
    const float e_hi = (lane <= 16) ? __expf(z_hi - m) : 0.0f;
    float s = e_lo + e_hi;
    #pragma unroll
    for (int off = 16; off; off >>= 1) s += __shfl_xor(s, off, 32);

    const float m50 = fmaxf(m, zext);
    const float s50 = s * __expf(m - m50) + __expf(zext - m50);

    const float inv = 1.0f / s;
    const size_t aBase = ((size_t)b * TT + t) * RR;
    alpha_out[aBase + lane] = e_lo * inv;                 // rows 0..31
    if (lane <= 16)
        alpha_out[aBase + 32 + lane] = e_hi * inv;        // rows 32..48
    if (lane == 0)
        beta_out[(size_t)b * TT + t] = __expf(zext - m50) / s50;
}

// ---------------------------------------------------------------------------
// Kernel 3: c_t = alpha(256x49) x V(49x512) per batch via WMMA f32 16x16x4,
// fused with c_hat = beta*s + (1-beta)*c_t (c_t never hits memory).
// alpha zero-padded to K=52 in LDS (52-float pitch: conflict-free A reads);
// V row clamped to 48 for pad lanes (alpha=0 there => exact, no OOB).
// ---------------------------------------------------------------------------
#define KPAD 52

__global__ __launch_bounds__(256) void attn_context_kernel(
    const float* __restrict__ V,      // [B][49][512]
    const float* __restrict__ s_t,    // [B][256][512]
    const float* __restrict__ alpha,  // [B][256][49]
    const float* __restrict__ beta,   // [B][256]
    float* __restrict__ chat)         // [B][256][512]
{
    __shared__ float aL[64 * KPAD];

    const int b     = blockIdx.x >> 2;
    const int tBase = (blockIdx.x & 3) << 6;     // 0,64,128,192
    const int tid   = threadIdx.x;
    const int wave  = tid >> 5;
    const int lane  = tid & 31;

    for (int i = tid; i < 64 * KPAD; i += 256) {
        const int row = i / KPAD, k = i % KPAD;
        aL[i] = (k < RR) ? alpha[((size_t)b * TT + tBase + row) * RR + k] : 0.0f;
    }
    __syncthreads();

    const int n  = lane & 15;
    const int kh = (lane >> 4) << 1;
    const float* __restrict__ Vb = V + (size_t)b * RR * HH;

    for (int tile = wave; tile < 128; tile += 8) {
        const int mt = tile >> 5;                // 0..3  (16 t-rows each)
        const int nt = tile & 31;                // 0..31 (16 h-cols each)
        const int mloc = (mt << 4) + n;
        const int h    = (nt << 4) + n;

        v8f acc = {};
        for (int k0 = 0; k0 < KPAD; k0 += 4) {
            const int k = k0 + kh;
            v2f a = *(const v2f*)(&aL[mloc * KPAD + k]);
            const int r0 = (k     < RR) ? k     : (RR - 1);
            const int r1 = (k + 1 < RR) ? k + 1 : (RR - 1);
            v2f bv = { Vb[(size_t)r0 * HH + h], Vb[(size_t)r1 * HH + h] };
            acc = WMMA_F32X4(a, bv, acc);
        }

        const int trow = tBase + (mt << 4) + ((lane >> 4) << 3);
        #pragma unroll
        for (int i = 0; i < 8; ++i) {
            const int t = trow + i;
            const float bt = beta[(size_t)b * TT + t];
            const size_t idx = ((size_t)b * TT + t) * HH + h;
            chat[idx] = bt * s_t[idx] + (1.0f - bt) * acc[i];
        }
    }
}

// ---------------------------------------------------------------------------
extern "C" void kernel_launch(void* const* d_in, const int* in_sizes, int n_in,
                              void* d_out, int out_size, void* d_ws, size_t ws_size,
                              hipStream_t stream)
{
    const float* V   = (const float*)d_in[0];
    const float* h_t = (const float*)d_in[1];
    const float* s_t = (const float*)d_in[2];
    const float* Wv  = (const float*)d_in[3];
    const float* Wg  = (const float*)d_in[4];
    const float* Wh  = (const float*)d_in[5];
    const float* Ws  = (const float*)d_in[6];

    float* out = (float*)d_out;
    float* chat  = out;                                   // B*T*H
    float* alpha = out + (size_t)BB * TT * HH;            // B*T*R
    float* beta  = alpha + (size_t)BB * TT * RR;          // B*T

    float* ws = (float*)d_ws;
    float* gv = ws;                                       // B*R*K
    float* gh = gv + (size_t)BB * RR * KK;                // B*T*K
    float* gs = gh + (size_t)BB * TT * KK;                // B*T*K

    // Projections: (M x 512) x (64 x 512)^T, one M-tile per wave, 4 waves/block.
    proj_gemm_kernel<<<(BB * RR) / 64, 128, 0, stream>>>(V,   Wv, gv);  // M=3136
    proj_gemm_kernel<<<(BB * TT) / 64, 128, 0, stream>>>(h_t, Wg, gh);  // M=16384
    proj_gemm_kernel<<<(BB * TT) / 64, 128, 0, stream>>>(s_t, Ws, gs);  // M=16384

    // Fused tanh-score + register softmax: 8 t per block.
    attn_scores_kernel<<<BB * (TT / 8), 256, 0, stream>>>(gv, gh, gs, Wh,
                                                          alpha, beta);

    // Fused context GEMM + gating: 4 blocks per batch (64 t-rows each).
    attn_context_kernel<<<BB * (TT / 64), 256, 0, stream>>>(V, s_t, alpha,
                                                            beta, chat);
}